// var_loss_70952859730214
// MI455X (gfx1250) — compile-verified
//
#include <hip/hip_runtime.h>
#include <math.h>

typedef __attribute__((ext_vector_type(2))) float v2f;
typedef __attribute__((ext_vector_type(8))) float v8f;

#define MAXL     1024
#define BATCHES  8
#define CHANNELS 64

// ---------------------------------------------------------------------------
// Kernel 0: zero the (sum,count) bin pairs in workspace. L read from device.
// ---------------------------------------------------------------------------
__global__ void vl_zero_bins(float* __restrict__ bins, const int* __restrict__ pL) {
    const int n = 2 * BATCHES * (*pL);
    for (int i = blockIdx.x * blockDim.x + threadIdx.x; i < n;
         i += gridDim.x * blockDim.x)
        bins[i] = 0.0f;
}

// ---------------------------------------------------------------------------
// Kernel 1: main streaming pass. Each wave handles a 16-point tile; channel
// reduction (S1, S2) is done with V_WMMA_F32_16X16X4_F32 using an all-ones
// A matrix (column-sum; invariant to B's K-row lane permutation).
// Requires N % 128 == 0 (true for N = 65536).
// ---------------------------------------------------------------------------
__global__ __launch_bounds__(256) void vl_main(const float* __restrict__ pred,
                                               const int*   __restrict__ label,
                                               const int*   __restrict__ pL,
                                               float*       __restrict__ bins,
                                               int N) {
    const int L = *pL;                      // uniform scalar load
    const bool useLds = (L <= MAXL);        // uniform branch

    __shared__ float lsum[MAXL];
    __shared__ float lcnt[MAXL];
    const int tid = threadIdx.x;

    if (useLds) {
        for (int i = tid; i < L; i += blockDim.x) { lsum[i] = 0.0f; lcnt[i] = 0.0f; }
        __syncthreads();
    }

    const int wave = tid >> 5;
    const int lane = tid & 31;
    const int hh   = lane >> 4;             // lane half -> channel pair select
    const int nl   = lane & 15;             // point index within tile
    const int b    = blockIdx.y;
    const int n0   = (blockIdx.x * 8 + wave) * 16;
    const int np   = n0 + nl;

    const float* base = pred + (size_t)b * CHANNELS * N + np;

    // Phase 1: issue all 32 loads per lane (2 channels per 4-channel K chunk).
    float xa[16], xb[16];
#pragma unroll
    for (int k = 0; k < 16; ++k) {
        const float* p = base + (size_t)(4 * k + 2 * hh) * N;
        xa[k] = p[0];
        xb[k] = p[(size_t)N];
    }

    // Phase 2: WMMA column-sum with all-ones A.
    // D[m][n] = sum_k B[k][n] for every m -> every lane's acc[0] holds the
    // reduction for point n0 + (lane & 15).
    const v2f aOnes = {1.0f, 1.0f};
    v8f acc1 = {0.f, 0.f, 0.f, 0.f, 0.f, 0.f, 0.f, 0.f};   // sum x
    v8f acc2 = {0.f, 0.f, 0.f, 0.f, 0.f, 0.f, 0.f, 0.f};   // sum x^2
#pragma unroll
    for (int k = 0; k < 16; ++k) {
        v2f bv = {xa[k], xb[k]};
        v2f bs = {xa[k] * xa[k], xb[k] * xb[k]};
        acc1 = __builtin_amdgcn_wmma_f32_16x16x4_f32(false, aOnes, false, bv,
                                                     (short)0, acc1, false, false);
        acc2 = __builtin_amdgcn_wmma_f32_16x16x4_f32(false, aOnes, false, bs,
                                                     (short)0, acc2, false, false);
    }

    // Lanes 0-15 commit one point each.
    if (lane < 16) {
        const float S1   = acc1[0];
        const float S2   = acc2[0];
        const float var  = S2 - S1 * S1 * (1.0f / (float)CHANNELS);
        const float dist = sqrtf(fmaxf(var, 0.0f));
        const float t    = fmaxf(dist - 0.5f, 0.0f);
        const float r    = t * t;
        const int   l    = label[(size_t)b * N + np];
        if (useLds) {
            atomicAdd(&lsum[l], r);
            atomicAdd(&lcnt[l], 1.0f);
        } else {
            atomicAdd(&bins[2 * ((size_t)b * L + l) + 0], r);
            atomicAdd(&bins[2 * ((size_t)b * L + l) + 1], 1.0f);
        }
    }

    if (useLds) {
        __syncthreads();
        for (int i = tid; i < L; i += blockDim.x) {
            const float c = lcnt[i];
            if (c != 0.0f) {
                atomicAdd(&bins[2 * ((size_t)b * L + i) + 0], lsum[i]);
                atomicAdd(&bins[2 * ((size_t)b * L + i) + 1], c);
            }
        }
    }
}

// ---------------------------------------------------------------------------
// Kernel 2: finalize 400 bins -> scalar. One block.
// per_group = sum/count (present only); per_batch = sum(per_group)/n_unique.
// ---------------------------------------------------------------------------
__global__ void vl_finalize(const float* __restrict__ bins,
                            const int*   __restrict__ pL,
                            float*       __restrict__ out) {
    const int L = *pL;
    __shared__ float gsum[BATCHES];
    __shared__ float gcnt[BATCHES];
    const int tid = threadIdx.x;
    if (tid < BATCHES) { gsum[tid] = 0.0f; gcnt[tid] = 0.0f; }
    __syncthreads();
    for (int i = tid; i < BATCHES * L; i += blockDim.x) {
        const int   b = i / L;
        const float s = bins[2 * i + 0];
        const float c = bins[2 * i + 1];
        if (c > 0.0f) {
            atomicAdd(&gsum[b], s / c);
            atomicAdd(&gcnt[b], 1.0f);
        }
    }
    __syncthreads();
    if (tid == 0) {
        float tot = 0.0f;
        for (int b = 0; b < BATCHES; ++b) tot += gsum[b] / gcnt[b];
        out[0] = tot;
    }
}

// ---------------------------------------------------------------------------
extern "C" void kernel_launch(void* const* d_in, const int* in_sizes, int n_in,
                              void* d_out, int out_size, void* d_ws, size_t ws_size,
                              hipStream_t stream) {
    const float* pred  = (const float*)d_in[0];
    const int*   label = (const int*)d_in[1];
    const int*   pL    = (const int*)d_in[2];
    float*       out   = (float*)d_out;
    float*       bins  = (float*)d_ws;

    const int BN = in_sizes[1];        // B * N
    const int B  = BATCHES;            // fixed by reference setup
    const int N  = BN / B;             // 65536

    vl_zero_bins<<<8, 256, 0, stream>>>(bins, pL);

    dim3 grid(N / 128, B);             // 8 waves/block * 16 points/wave = 128
    vl_main<<<grid, 256, 0, stream>>>(pred, label, pL, bins, N);

    vl_finalize<<<1, 256, 0, stream>>>(bins, pL, out);
}